// VNREstimator_59562606461181
// MI455X (gfx1250) — compile-verified
//
#include <hip/hip_runtime.h>
#include <hip/hip_bf16.h>

#define B_   16
#define F_   257
#define FP   260          // F padded to multiple of 4 (WMMA K granularity)
#define T_   8000
#define NB_  64
#define TT   32           // time-tile width (2 x WMMA N)
#define NTILES (T_ / TT)  // 250
#define NPROD_WAVES 9     // 288 lanes cover 257 chains + 3 zero-pad lanes
#define NCONS_WAVES 4     // one 16-row band tile each (4*16 = 64 bands)
#define NTHREADS ((NPROD_WAVES + NCONS_WAVES) * 32)  // 416

typedef __attribute__((ext_vector_type(2))) float v2f;
typedef __attribute__((ext_vector_type(4))) float v4f;
typedef __attribute__((ext_vector_type(8))) float v8f;

__device__ __forceinline__ float fast_tanh(float x) {
#if __has_builtin(__builtin_amdgcn_tanhf)
    return __builtin_amdgcn_tanhf(x);          // native V_TANH_F32 (CDNA5 trans op)
#else
    float r;
    asm volatile("v_tanh_f32 %0, %1\n\tv_nop"  // v_nop covers trans-op result hazard
                 : "=v"(r) : "v"(x));
    return r;
#endif
}

// Producer: advance TT recurrence steps, emit vnr tile column f into LDS buf[t][f].
// RISE_GE_FALL selects max-combine (rise>=fall) vs min-combine: the data-dependent
// alpha select is equivalent to picking the larger/smaller of the two fma results.
template <bool RISE_GE_FALL>
__device__ __forceinline__ void produce_tile(const float* __restrict__ row, int tile,
                                             float& nf, float rise, float fall,
                                             float floorm, float ns,
                                             float* __restrict__ buf,
                                             int f, bool active, bool zeroer)
{
    if (active) {
        const v4f* p = reinterpret_cast<const v4f*>(row + tile * TT);
        float xs[TT];
        #pragma unroll
        for (int i = 0; i < TT / 4; ++i) {
            v4f m = p[i];
            xs[4*i+0] = m[0]; xs[4*i+1] = m[1]; xs[4*i+2] = m[2]; xs[4*i+3] = m[3];
        }
        #pragma unroll
        for (int t = 0; t < TT; ++t) {
            float x  = xs[t];
            float d  = x - nf;
            float a1 = fmaf(rise, d, nf);
            float a2 = fmaf(fall, d, nf);
            if (RISE_GE_FALL)
                nf = fmaxf(fmaxf(a1, a2), floorm);      // v_max3_f32: 3-deep chain
            else
                nf = fmaxf(fminf(a1, a2), floorm);
            buf[t * FP + f] = x * __builtin_amdgcn_rcpf(fmaf(ns, nf, 1e-8f));
        }
    } else if (zeroer) {                                // K-padding rows 257..259
        #pragma unroll
        for (int t = 0; t < TT; ++t) buf[t * FP + f] = 0.0f;
    }
}

__global__ __launch_bounds__(NTHREADS, 1)
void VNREstimator_kernel(const float* __restrict__ mag,
                         const float* __restrict__ fb,
                         const float* __restrict__ ns_p,
                         const float* __restrict__ rr_p,
                         const float* __restrict__ rf_p,
                         float* __restrict__ out)
{
    __shared__ float lds_fb[NB_ * FP];        // filterbank, K-padded with zeros
    __shared__ float tilebuf[2][TT * FP];     // double-buffered vnr tile [t][f]

    const int b    = blockIdx.x;
    const int tid  = threadIdx.x;
    const int wid  = tid >> 5;
    const int lane = tid & 31;

    // Stage freq_fb into LDS (all waves cooperate), zero the pad columns.
    for (int i = tid; i < NB_ * FP; i += NTHREADS) {
        int f = i % FP;
        int n = i / FP;
        lds_fb[i] = (f < F_) ? fb[n * F_ + f] : 0.0f;
    }

    const float rise = 1.0f / (1.0f + expf(-rr_p[0]));
    const float fall = 1.0f / (1.0f + expf(-rf_p[0]));
    const float ns   = fabsf(ns_p[0]);

    if (wid < NPROD_WAVES) {
        // ---------------- producer waves: the sequential noise-floor scan ----------
        const int  f      = tid;              // 0..287
        const bool active = (f < F_);
        const bool zeroer = (f >= F_ && f < FP);
        const int  fc     = active ? f : (F_ - 1);
        const float* row  = mag + ((size_t)b * F_ + fc) * T_;

        float nf = 0.0f, floorm = 0.0f;
        if (active) {
            float mn = row[0];
            #pragma unroll
            for (int t = 1; t < 20; ++t) mn = fminf(mn, row[t]);
            mn = fmaxf(mn, 1e-5f);            // clip(init_min, 1e-5)
            nf = mn; floorm = 0.5f * mn;
        }

        if (rise >= fall) {                   // wave-uniform branch
            produce_tile<true>(row, 0, nf, rise, fall, floorm, ns, tilebuf[0], f, active, zeroer);
            __syncthreads();
            for (int tile = 0; tile < NTILES; ++tile) {
                if (tile + 1 < NTILES)
                    produce_tile<true>(row, tile + 1, nf, rise, fall, floorm, ns,
                                       tilebuf[(tile + 1) & 1], f, active, zeroer);
                __syncthreads();              // consumer drains buf[tile&1] meanwhile
            }
        } else {
            produce_tile<false>(row, 0, nf, rise, fall, floorm, ns, tilebuf[0], f, active, zeroer);
            __syncthreads();
            for (int tile = 0; tile < NTILES; ++tile) {
                if (tile + 1 < NTILES)
                    produce_tile<false>(row, tile + 1, nf, rise, fall, floorm, ns,
                                        tilebuf[(tile + 1) & 1], f, active, zeroer);
                __syncthreads();
            }
        }
    } else {
        // ---------------- consumer waves: WMMA band projection + tanh -------------
        const int w       = wid - NPROD_WAVES;   // 0..3 -> bands 16w..16w+15
        const int l15     = lane & 15;
        const int halfsel = (lane < 16) ? 0 : 1; // K pair {0,1} vs {2,3}
        // A fragment base: fb row (16w + l15), K offset 2*halfsel
        const float* arow = lds_fb + (w * 16 + l15) * FP + 2 * halfsel;

        __syncthreads();                      // wait for fb staging + tile 0
        for (int tile = 0; tile < NTILES; ++tile) {
            // B fragments: time cols l15 (c0) and l15+16 (c1), shared A fragment
            const float* brow0 = tilebuf[tile & 1] + l15 * FP + 2 * halfsel;
            const float* brow1 = brow0 + 16 * FP;
            v8f c0 = {}, c1 = {};
            #pragma unroll 5
            for (int kk = 0; kk < FP / 4; ++kk) {   // 65 K-steps of 4
                v2f a  = *reinterpret_cast<const v2f*>(arow  + 4 * kk);
                v2f b0 = *reinterpret_cast<const v2f*>(brow0 + 4 * kk);
                v2f b1 = *reinterpret_cast<const v2f*>(brow1 + 4 * kk);
                c0 = __builtin_amdgcn_wmma_f32_16x16x4_f32(false, a, false, b0,
                                                           (short)0, c0, false, false);
                c1 = __builtin_amdgcn_wmma_f32_16x16x4_f32(false, a, false, b1,
                                                           (short)0, c1, false, false);
            }
            // C/D layout: VGPR r -> band row (w*16 + r + 8*halfsel), time col = l15
            const int tcol = tile * TT + l15;
            #pragma unroll
            for (int r = 0; r < 8; ++r) {
                int n = w * 16 + r + 8 * halfsel;
                float* orow = out + ((size_t)b * NB_ + n) * T_;
                float v0 = fast_tanh(c0[r] * 0.1f);
                float v1 = fast_tanh(c1[r] * 0.1f);
                v0 = (v0 == v0) ? v0 : 0.0f;  // nan_to_num (tanh output in [-1,1])
                v1 = (v1 == v1) ? v1 : 0.0f;
                orow[tcol]      = v0;
                orow[tcol + 16] = v1;
            }
            __syncthreads();
        }
    }
}

extern "C" void kernel_launch(void* const* d_in, const int* in_sizes, int n_in,
                              void* d_out, int out_size, void* d_ws, size_t ws_size,
                              hipStream_t stream) {
    const float* mag = (const float*)d_in[0];
    const float* fb  = (const float*)d_in[1];
    const float* ns  = (const float*)d_in[2];
    const float* rr  = (const float*)d_in[3];
    const float* rf  = (const float*)d_in[4];
    float* out = (float*)d_out;
    hipLaunchKernelGGL(VNREstimator_kernel, dim3(B_), dim3(NTHREADS), 0, stream,
                       mag, fb, ns, rr, rf, out);
}